// SelfAttention_13073880449780
// MI455X (gfx1250) — compile-verified
//
#include <hip/hip_runtime.h>

// MI455X / gfx1250, wave32. fp32 matrix path via V_WMMA_F32_16X16X4_F32.
// Workspace layout (floats): xn[8388608] | qkv[25165824] | vT[8388608] = 160 MB.

typedef __attribute__((ext_vector_type(2))) float v2f;
typedef __attribute__((ext_vector_type(8))) float v8f;

#define NB 16
#define NC 512
#define NHW 1024
#define NG 32
#define NH 4
#define ND 128
#define GN_EPS 1e-5f
#define ATT_SCALE 0.08838834764831845f  // 128^-0.5

static __device__ __forceinline__ v8f wmma4(v2f a, v2f b, v8f c) {
  // D = A(16x4 f32) * B(4x16 f32) + C(16x16 f32)
  return __builtin_amdgcn_wmma_f32_16x16x4_f32(false, a, false, b, (short)0, c,
                                               false, false);
}

// ---------------- GroupNorm ----------------
// one block per (b, group): 16 channels x 1024 px = 16384 floats
__global__ __launch_bounds__(256) void gn_kernel(const float* __restrict__ x,
                                                 const float* __restrict__ w,
                                                 const float* __restrict__ bia,
                                                 float* __restrict__ xn) {
  const int b = blockIdx.x >> 5;
  const int g = blockIdx.x & 31;
  const size_t base = ((size_t)b * NC + (size_t)g * 16) * NHW;
  const int tid = threadIdx.x;
  float sum = 0.f, sq = 0.f;
  for (int i = tid; i < 16 * NHW; i += 256) {
    float v = x[base + i];
    sum += v;
    sq += v * v;
  }
  for (int m = 16; m >= 1; m >>= 1) {
    sum += __shfl_xor(sum, m, 32);
    sq += __shfl_xor(sq, m, 32);
  }
  __shared__ float rs[8], rq[8], stat[2];
  const int wv = tid >> 5;
  if ((tid & 31) == 0) { rs[wv] = sum; rq[wv] = sq; }
  __syncthreads();
  if (tid == 0) {
    float ts = 0.f, tq = 0.f;
    for (int i = 0; i < 8; ++i) { ts += rs[i]; tq += rq[i]; }
    float mean = ts * (1.f / 16384.f);
    float var = tq * (1.f / 16384.f) - mean * mean;
    stat[0] = mean;
    stat[1] = rsqrtf(var + GN_EPS);
  }
  __syncthreads();
  const float mean = stat[0], rstd = stat[1];
  for (int i = tid; i < 16 * NHW; i += 256) {
    int c = (g << 4) + (i >> 10);
    xn[base + i] = (x[base + i] - mean) * rstd * w[c] + bia[c];
  }
}

// ---------------- fp32 WMMA GEMM, K=512, N=1024, batched over blockIdx.y ----
// Y[b][m][n] = sum_k A[m][k] * Bmat[b][k][n]  (+ bias[m] + resid[b][m][n])
__global__ __launch_bounds__(256) void gemm512_kernel(
    const float* __restrict__ A, const float* __restrict__ Bmat,
    float* __restrict__ Y, long bStrideB, long bStrideY,
    const float* __restrict__ bias, const float* __restrict__ resid) {
  __shared__ float ldsA[16 * 516];  // stride 516: 516 % 64 == 4 -> no bank conflicts
  const int m0 = blockIdx.x << 4;
  const int b = blockIdx.y;
  const int tid = threadIdx.x;
  for (int i = tid; i < 8192; i += 256) {
    int row = i >> 9, k = i & 511;
    ldsA[row * 516 + k] = A[(size_t)(m0 + row) * 512 + k];
  }
  __syncthreads();
  const int lane = tid & 31, wv = tid >> 5;
  const int lm = lane & 15, hi = lane >> 4;
  const float* Bb = Bmat + (size_t)b * bStrideB;
  const int n0 = wv << 7;  // 8 waves x 128 columns = full N
  v8f acc[8] = {};
  for (int k0 = 0; k0 < 512; k0 += 4) {
    const int kr = k0 + 2 * hi;  // A lane holds (M=lm, K=kr..kr+1)
    v2f a;
    a.x = ldsA[lm * 516 + kr];
    a.y = ldsA[lm * 516 + kr + 1];
    const float* bp = Bb + (size_t)kr * NHW + n0 + lm;
#pragma unroll
    for (int nc = 0; nc < 8; ++nc) {
      v2f bf;
      bf.x = bp[nc * 16];          // B lane holds (K=kr,   N=n0+nc*16+lm)
      bf.y = bp[nc * 16 + NHW];    //              (K=kr+1, ...)
      acc[nc] = wmma4(a, bf, acc[nc]);
    }
  }
  float* Yb = Y + (size_t)b * bStrideY;
#pragma unroll
  for (int nc = 0; nc < 8; ++nc) {
#pragma unroll
    for (int r = 0; r < 8; ++r) {
      const int row = m0 + r + 8 * hi;  // C/D layout: M = r + 8*(lane/16)
      const int col = n0 + (nc << 4) + lm;
      float v = acc[nc][r];
      if (bias)
        v += bias[row] + resid[(size_t)b * bStrideY + (size_t)row * NHW + col];
      Yb[(size_t)row * NHW + col] = v;
    }
  }
}

// ---------------- V transpose: v[b][h][d][j] -> vT[b][h][j][d] ------------
__global__ void vtrans_kernel(const float* __restrict__ qkv,
                              float* __restrict__ vt) {
  __shared__ float t[16][17];
  const int bh = blockIdx.z, bq = bh >> 2, h = bh & 3;
  const int j0 = blockIdx.x << 4, d0 = blockIdx.y << 4;
  const int tx = threadIdx.x, ty = threadIdx.y;
  t[ty][tx] =
      qkv[((size_t)bq * 1536 + 1024 + h * ND + d0 + ty) * NHW + j0 + tx];
  __syncthreads();
  vt[((size_t)bh * NHW + j0 + ty) * ND + d0 + tx] = t[tx][ty];
}

// ---------------- Flash attention (fp32 WMMA, online softmax) --------------
// one wave = 16 queries; workgroup = 8 waves = 128 queries of one (b,h).
__global__ __launch_bounds__(256) void attn_kernel(const float* __restrict__ qkv,
                                                   const float* __restrict__ vt,
                                                   float* __restrict__ att) {
  __shared__ float ldsK[128 * 16];     // K tile [d][j]
  __shared__ float ldsV[16 * 132];     // V^T tile [j][d], stride 132 (132%64==4)
  __shared__ float ldsP[8][16 * 17];   // per-wave P layout shuffle (17%64 coprime)
  const int bh = blockIdx.x >> 3, iblk = blockIdx.x & 7;
  const int b = bh >> 2, h = bh & 3;
  const int tid = threadIdx.x, wv = tid >> 5, lane = tid & 31;
  const int lm = lane & 15, hi = lane >> 4;
  const int i0 = (iblk << 7) + (wv << 4);
  const float* qp = qkv + ((size_t)b * 1536 + h * ND) * NHW;
  const float* kp = qkv + ((size_t)b * 1536 + 512 + h * ND) * NHW;
  const float* vp = vt + (size_t)bh * NHW * ND;
  // preload all Q A-fragments: A[m=i, k=d], lane holds (M=lm, K=kr..kr+1)
  v2f qf[32];
#pragma unroll
  for (int ks = 0; ks < 32; ++ks) {
    const int kr = ks * 4 + 2 * hi;
    qf[ks].x = qp[(size_t)kr * NHW + i0 + lm];
    qf[ks].y = qp[(size_t)(kr + 1) * NHW + i0 + lm];
  }
  v8f O[8] = {};  // O tile [i x 128d]: O[nc][r] = (i=i0+r+8*hi, d=nc*16+lm)
  float mst[8], lst[8];
#pragma unroll
  for (int r = 0; r < 8; ++r) { mst[r] = -3.0e38f; lst[r] = 0.f; }
  float* myP = &ldsP[wv][0];

  for (int jt = 0; jt < 64; ++jt) {
    const int j0 = jt << 4;
    __syncthreads();
    for (int i = tid; i < 2048; i += 256)  // stage K [128x16]
      ldsK[i] = kp[(size_t)(i >> 4) * NHW + j0 + (i & 15)];
    for (int i = tid; i < 2048; i += 256)  // stage V^T [16x128]
      ldsV[(i >> 7) * 132 + (i & 127)] = vp[(size_t)(j0 + (i >> 7)) * ND + (i & 127)];
    __syncthreads();
    // S = Q^T K (16x16), K-dim = d = 128
    v8f s = {};
#pragma unroll
    for (int ks = 0; ks < 32; ++ks) {
      const int kr = ks * 4 + 2 * hi;
      v2f bf;
      bf.x = ldsK[kr * 16 + lm];
      bf.y = ldsK[(kr + 1) * 16 + lm];
      s = wmma4(qf[ks], bf, s);
    }
    // online softmax: row i = r+8*hi lives in the 16 lanes of this half-wave
    float alpha[8], p[8];
#pragma unroll
    for (int r = 0; r < 8; ++r) {
      float sv = s[r] * ATT_SCALE;
      float tm = sv;
      tm = fmaxf(tm, __shfl_xor(tm, 1, 16));
      tm = fmaxf(tm, __shfl_xor(tm, 2, 16));
      tm = fmaxf(tm, __shfl_xor(tm, 4, 16));
      tm = fmaxf(tm, __shfl_xor(tm, 8, 16));
      float mn = fmaxf(mst[r], tm);
      alpha[r] = __expf(mst[r] - mn);
      float pv = __expf(sv - mn);
      float ts = pv;
      ts += __shfl_xor(ts, 1, 16);
      ts += __shfl_xor(ts, 2, 16);
      ts += __shfl_xor(ts, 4, 16);
      ts += __shfl_xor(ts, 8, 16);
      lst[r] = lst[r] * alpha[r] + ts;
      mst[r] = mn;
      p[r] = pv;
    }
#pragma unroll
    for (int nc = 0; nc < 8; ++nc)
#pragma unroll
      for (int r = 0; r < 8; ++r) O[nc][r] *= alpha[r];
    // C/D-layout P -> row-major LDS slab (wave-private, no barrier needed)
#pragma unroll
    for (int r = 0; r < 8; ++r) myP[(r + 8 * hi) * 17 + lm] = p[r];
    // O += P @ V^T : A = P [16i x 16j], B = V^T [16j x 16d] per n-chunk
#pragma unroll
    for (int ks = 0; ks < 4; ++ks) {
      const int kr = ks * 4 + 2 * hi;
      v2f af;
      af.x = myP[lm * 17 + kr];
      af.y = myP[lm * 17 + kr + 1];
#pragma unroll
      for (int nc = 0; nc < 8; ++nc) {
        v2f bf;
        bf.x = ldsV[kr * 132 + (nc << 4) + lm];
        bf.y = ldsV[(kr + 1) * 132 + (nc << 4) + lm];
        O[nc] = wmma4(af, bf, O[nc]);
      }
    }
  }
  // normalize + write att[b][h*128+d][i]
  float* ap = att + ((size_t)b * NC + h * ND) * NHW;
#pragma unroll
  for (int r = 0; r < 8; ++r) {
    const float inv = 1.f / lst[r];
#pragma unroll
    for (int nc = 0; nc < 8; ++nc)
      ap[(size_t)((nc << 4) + lm) * NHW + i0 + r + 8 * hi] = O[nc][r] * inv;
  }
}

extern "C" void kernel_launch(void* const* d_in, const int* in_sizes, int n_in,
                              void* d_out, int out_size, void* d_ws,
                              size_t ws_size, hipStream_t stream) {
  (void)in_sizes; (void)n_in; (void)out_size; (void)ws_size;
  const float* x = (const float*)d_in[0];
  const float* gnw = (const float*)d_in[1];
  const float* gnb = (const float*)d_in[2];
  const float* wqkv = (const float*)d_in[3];
  const float* wout = (const float*)d_in[4];
  const float* bout = (const float*)d_in[5];
  float* out = (float*)d_out;
  float* ws = (float*)d_ws;
  float* xn = ws;                 // 8,388,608 f (reused as attention output)
  float* qkv = ws + 8388608;      // 25,165,824 f
  float* vt = ws + 33554432;      // 8,388,608 f

  gn_kernel<<<NB * NG, 256, 0, stream>>>(x, gnw, gnb, xn);
  gemm512_kernel<<<dim3(96, NB), 256, 0, stream>>>(wqkv, xn, qkv, 524288L,
                                                   1572864L, nullptr, nullptr);
  vtrans_kernel<<<dim3(64, 8, NB * NH), dim3(16, 16), 0, stream>>>(qkv, vt);
  attn_kernel<<<NB * NH * 8, 256, 0, stream>>>(qkv, vt, xn);
  gemm512_kernel<<<dim3(32, NB), 256, 0, stream>>>(wout, xn, out, 524288L,
                                                   524288L, bout, x);
}